// GIN_22084721836226
// MI455X (gfx1250) — compile-verified
//
#include <hip/hip_runtime.h>

#define N_NODES 100000
#define N_EDGES 1600000
#define N_GRAPH 2048
#define DIN     11
#define DH      128
#define BN_EPS  1e-5f

typedef __attribute__((ext_vector_type(2))) float v2f;
typedef __attribute__((ext_vector_type(8))) float v8f;
typedef __attribute__((__vector_size__(4 * sizeof(int)))) int v4i;
typedef __attribute__((address_space(1))) v4i* gptr_v4i; // global (prints as __device__)
typedef __attribute__((address_space(3))) v4i* lptr_v4i; // LDS

// ---------------------------------------------------------- async helpers
#if defined(__has_builtin)
#if __has_builtin(__builtin_amdgcn_global_load_async_to_lds_b128)
#define HAVE_ASYNC_LDS_BUILTIN 1
#endif
#if __has_builtin(__builtin_amdgcn_s_wait_asynccnt)
#define HAVE_WAIT_ASYNC_BUILTIN 1
#endif
#endif

__device__ __forceinline__ void async_copy_32B(const float* g, float* l) {
#if defined(HAVE_ASYNC_LDS_BUILTIN)
    __builtin_amdgcn_global_load_async_to_lds_b128((gptr_v4i)(g),
                                                   (lptr_v4i)(l), 0, 0);
    __builtin_amdgcn_global_load_async_to_lds_b128((gptr_v4i)(g + 4),
                                                   (lptr_v4i)(l + 4), 0, 0);
#else
    unsigned l0 = (unsigned)(__UINTPTR_TYPE__)(__attribute__((address_space(3))) float*)(l);
    asm volatile("global_load_async_to_lds_b128 %0, %1, off"
                 :: "v"(l0), "v"(g) : "memory");
    asm volatile("global_load_async_to_lds_b128 %0, %1, off"
                 :: "v"(l0 + 16u), "v"(g + 4) : "memory");
#endif
}

__device__ __forceinline__ void wait_async0() {
#if defined(HAVE_WAIT_ASYNC_BUILTIN)
    __builtin_amdgcn_s_wait_asynccnt(0);
#else
    asm volatile("s_wait_asynccnt 0x0" ::: "memory");
#endif
}

// ---------------------------------------------------------------- utilities
__global__ void k_zero(float* __restrict__ p, long long n) {
    long long i = (long long)blockIdx.x * blockDim.x + threadIdx.x;
    if (i < n) p[i] = 0.0f;
}

__global__ void k_add_inplace(float* __restrict__ a, const float* __restrict__ b, long long n) {
    long long i = (long long)blockIdx.x * blockDim.x + threadIdx.x;
    if (i < n) a[i] += b[i];
}

// Fold BatchNorm (eval) into per-feature scale/shift: y = x*scale + shift
__global__ void k_bnfold(const float* __restrict__ g, const float* __restrict__ be,
                         const float* __restrict__ m, const float* __restrict__ v,
                         float* __restrict__ scale, float* __restrict__ shift) {
    int i = threadIdx.x; // 128 threads
    float s = g[i] * rsqrtf(v[i] + BN_EPS);
    scale[i] = s;
    shift[i] = be[i] - m[i] * s;
}

// ------------------------------------------------------- edge scatter (sum)
// agg[dst[e], f] += h[src[e], f]; F consecutive threads per edge (coalesced).
__global__ void k_scatter(const float* __restrict__ h, const int* __restrict__ src,
                          const int* __restrict__ dst, float* __restrict__ agg, int F) {
    long long i = (long long)blockIdx.x * blockDim.x + threadIdx.x;
    long long total = (long long)N_EDGES * F;
    if (i >= total) return;
    int e = (int)(i / F);
    int f = (int)(i % F);
    atomicAdd(&agg[(long long)dst[e] * F + f], h[(long long)src[e] * F + f]);
}

// ----------------------------------------- layer-1 first linear (K = 11)
__global__ void k_lin_l1(const float* __restrict__ x, const float* __restrict__ agg,
                         const float* __restrict__ W, const float* __restrict__ b,
                         const float* __restrict__ scale, const float* __restrict__ shift,
                         float* __restrict__ out) {
    long long i = (long long)blockIdx.x * blockDim.x + threadIdx.x;
    if (i >= (long long)N_NODES * DH) return;
    int n = (int)(i >> 7);
    int f = (int)(i & 127);
    float acc = b[f];
#pragma unroll
    for (int k = 0; k < DIN; ++k) {
        float t = x[n * DIN + k] + agg[n * DIN + k];
        acc = fmaf(t, W[k * DH + f], acc);
    }
    acc = fmaf(acc, scale[f], shift[f]);
    out[i] = fmaxf(acc, 0.0f);
}

// ------------------------------------------------ WMMA fp32 GEMM, K=N=128
// out = relu( (A @ W + bias) [* scale + shift] ), rows % 16 == 0.
// Block: 256 threads = 8 waves; wave -> one 16-col tile; block -> row_tiles
// consecutive 16-row tiles. B column-tile resident in 64 VGPRs per lane;
// A tiles double-buffered through LDS via async-to-LDS copies (ASYNCcnt).
// In-place (out == A) is safe: A is only read by the async copies of tiles
// >= t, stores only hit tiles < t (disjoint 16-row blocks).
template <bool BN>
__global__ __launch_bounds__(256) void k_gemm128(
        const float* A, const float* __restrict__ W,
        const float* __restrict__ bias, const float* __restrict__ scale,
        const float* __restrict__ shift, float* out, int row_tiles) {
    __shared__ float atile[2][16 * DH]; // 2 x 8KB

    const int tid  = threadIdx.x;
    const int wave = tid >> 5;
    const int lane = tid & 31;
    const int half = lane >> 4;  // 0: lanes 0-15, 1: lanes 16-31
    const int mn   = lane & 15;
    const int col  = wave * 16 + mn;
    const int tile0 = blockIdx.x * row_tiles;

    // B resident in registers: breg[kk] holds W[4*kk + 2*half + {0,1}][col]
    const float* bcol = W + (long long)(2 * half) * DH + col;
    v2f breg[32];
#pragma unroll
    for (int kk = 0; kk < 32; ++kk) {
        breg[kk].x = bcol[(long long)(4 * kk + 0) * DH];
        breg[kk].y = bcol[(long long)(4 * kk + 1) * DH];
    }

    const float bi = bias[col];
    const float sc = BN ? scale[col] : 1.0f;
    const float sh = BN ? shift[col] : 0.0f;

    // Prologue: async-stage tile 0 (a 16-row tile is 8KB contiguous in memory)
    async_copy_32B(A + (long long)tile0 * 16 * DH + tid * 8, &atile[0][tid * 8]);

    for (int t = 0; t < row_tiles; ++t) {
        wait_async0();       // this wave's portion of tile t landed in LDS
        __syncthreads();     // all waves' portions landed; prior buf reads done
        if (t + 1 < row_tiles)
            async_copy_32B(A + (long long)(tile0 + t + 1) * 16 * DH + tid * 8,
                           &atile[(t + 1) & 1][tid * 8]);

        // A element for lane: (row = mn, k = 4*kk + 2*half + {0,1})
        const float* ab = &atile[t & 1][mn * DH + 2 * half];
        v8f c = {0.f, 0.f, 0.f, 0.f, 0.f, 0.f, 0.f, 0.f};
#pragma unroll
        for (int kk = 0; kk < 32; ++kk) {
            v2f a = *(const v2f*)(ab + 4 * kk); // ds_load_b64
            c = __builtin_amdgcn_wmma_f32_16x16x4_f32(false, a, false, breg[kk],
                                                      (short)0, c, false, false);
        }

        const int row_base = (tile0 + t) * 16;
#pragma unroll
        for (int v = 0; v < 8; ++v) {
            float val = c[v] + bi;           // C/D layout: VGPR v -> M = v (+8 hi half)
            if (BN) val = fmaf(val, sc, sh);
            out[(long long)(row_base + v + 8 * half) * DH + col] = fmaxf(val, 0.0f);
        }
    }
}

// ------------------------------------------------------------- pooling
__global__ void k_pool(const float* __restrict__ h, const int* __restrict__ batch,
                       float* __restrict__ pooled) {
    long long i = (long long)blockIdx.x * blockDim.x + threadIdx.x;
    if (i >= (long long)N_NODES * DH) return;
    int n = (int)(i >> 7);
    int f = (int)(i & 127);
    atomicAdd(&pooled[(long long)batch[n] * DH + f], h[i]);
}

// ------------------------------------------------------------- final head
__global__ void k_lin2(const float* __restrict__ p, const float* __restrict__ w,
                       const float* __restrict__ b, float* __restrict__ out) {
    int g = blockIdx.x * blockDim.x + threadIdx.x;
    if (g >= N_GRAPH) return;
    float acc = b[0];
#pragma unroll 8
    for (int k = 0; k < DH; ++k) acc = fmaf(p[g * DH + k], w[k], acc);
    out[g] = acc;
}

// ---------------------------------------------------------------- driver
static inline unsigned cdiv(long long a, long long b) { return (unsigned)((a + b - 1) / b); }

extern "C" void kernel_launch(void* const* d_in, const int* in_sizes, int n_in,
                              void* d_out, int out_size, void* d_ws, size_t ws_size,
                              hipStream_t stream) {
    (void)in_sizes; (void)n_in; (void)out_size; (void)ws_size;

    const float* x     = (const float*)d_in[0];
    const int*   ei    = (const int*)d_in[1];
    const int*   src   = ei;
    const int*   dst   = ei + N_EDGES;
    const int*   batch = (const int*)d_in[2];

    // Per-layer params start at index 3, 8 tensors each: Wa, ba, g, be, m, v, Wb, bb
    const float* P[24];
    for (int i = 0; i < 24; ++i) P[i] = (const float*)d_in[3 + i];
    const float* lin1_W = (const float*)d_in[27];
    const float* lin1_b = (const float*)d_in[28];
    const float* lin2_W = (const float*)d_in[29];
    const float* lin2_b = (const float*)d_in[30];

    float* ws    = (float*)d_ws;
    float* bufH  = ws;                               // [N, 128] node features
    float* bufA  = ws + (size_t)N_NODES * DH;        // [N, 128] agg / pooled scratch
    float* scale = ws + 2 * (size_t)N_NODES * DH;    // [3,128]
    float* shift = scale + 3 * DH;                   // [3,128]
    float* outF  = (float*)d_out;                    // [G] fp32

    const long long NF = (long long)N_NODES * DH;
    const int TPB_N = 10;                            // 6250 tiles = 625 blocks x 10
    const int TPB_G = 8;                             // 128 tiles  = 16 blocks  x 8

    // Fold BN for the 3 layers
    for (int l = 0; l < 3; ++l)
        k_bnfold<<<1, DH, 0, stream>>>(P[8 * l + 2], P[8 * l + 3], P[8 * l + 4],
                                       P[8 * l + 5], scale + l * DH, shift + l * DH);

    // ---- Layer 1 (K = 11 first linear) ----
    {
        long long nA = (long long)N_NODES * DIN;
        k_zero<<<cdiv(nA, 256), 256, 0, stream>>>(bufA, nA);
        long long nS = (long long)N_EDGES * DIN;
        k_scatter<<<cdiv(nS, 256), 256, 0, stream>>>(x, src, dst, bufA, DIN);
        k_lin_l1<<<cdiv(NF, 256), 256, 0, stream>>>(x, bufA, P[0], P[1],
                                                    scale, shift, bufH);
        k_gemm128<false><<<(N_NODES / 16) / TPB_N, 256, 0, stream>>>(
            bufH, P[6], P[7], scale, shift, bufH, TPB_N);
    }

    // ---- Layers 2, 3 (full WMMA path) ----
    for (int l = 1; l < 3; ++l) {
        const float* Wa = P[8 * l + 0];
        const float* ba = P[8 * l + 1];
        const float* Wb = P[8 * l + 6];
        const float* bb = P[8 * l + 7];
        k_zero<<<cdiv(NF, 256), 256, 0, stream>>>(bufA, NF);
        long long nS = (long long)N_EDGES * DH;
        k_scatter<<<cdiv(nS, 256), 256, 0, stream>>>(bufH, src, dst, bufA, DH);
        k_add_inplace<<<cdiv(NF, 256), 256, 0, stream>>>(bufA, bufH, NF); // t = h + agg
        k_gemm128<true><<<(N_NODES / 16) / TPB_N, 256, 0, stream>>>(
            bufA, Wa, ba, scale + l * DH, shift + l * DH, bufH, TPB_N);
        k_gemm128<false><<<(N_NODES / 16) / TPB_N, 256, 0, stream>>>(
            bufH, Wb, bb, scale, shift, bufH, TPB_N);
    }

    // ---- Pool + head ----
    {
        long long nP = (long long)N_GRAPH * DH;
        k_zero<<<cdiv(nP, 256), 256, 0, stream>>>(bufA, nP);
        k_pool<<<cdiv(NF, 256), 256, 0, stream>>>(bufH, batch, bufA);
        k_gemm128<false><<<(N_GRAPH / 16) / TPB_G, 256, 0, stream>>>(
            bufA, lin1_W, lin1_b, scale, shift, bufA, TPB_G);
        k_lin2<<<cdiv(N_GRAPH, 256), 256, 0, stream>>>(bufA, lin2_W, lin2_b, outF);
    }
}